// KernelSmoothedIntegrator_89335319757298
// MI455X (gfx1250) — compile-verified
//
#include <hip/hip_runtime.h>
#include <hip/hip_bf16.h>

// Problem constants (from reference)
#define BB_ 4
#define TT_ 1024
#define KK_ 64
#define DD_ 1024
#define VV_ 32000
#define BT_ (BB_ * TT_)      // 4096 tokens
#define D2_ (2 * DD_)        // 2048

typedef __attribute__((ext_vector_type(2))) float        v2f;
typedef __attribute__((ext_vector_type(8))) float        v8f;
typedef __attribute__((ext_vector_type(4))) unsigned int v4u;
typedef __attribute__((ext_vector_type(8))) int          v8i;
typedef __attribute__((ext_vector_type(4))) int          v4i;

// ---------------------------------------------------------------------------
// Build a TDM descriptor for a 2-D fp32 tile (rows x 1024 cols, contiguous)
// and issue TENSOR_LOAD_TO_LDS. Descriptor bit layout per CDNA5 ISA §8.3/8.4:
//   group0: [1:0]=count=1, [63:32]=lds_addr, [120:64]=global_addr, [127:126]=2
//   group1: [17:16]=data_size(2=4B), [79:48]=tensor_dim0, [111:80]=tensor_dim1,
//           [127:112]=tile_dim0, [143:128]=tile_dim1, [207:160]=dim0_stride
// This toolchain exposes the 6-arg builtin: (g0, g1, g2, g3, g4, cpol).
// ---------------------------------------------------------------------------
__device__ __forceinline__ void tdm_load_rows_1024f(const float* gsrc,
                                                    unsigned lds_byte_addr,
                                                    unsigned rows) {
    unsigned long long ga = (unsigned long long)(uintptr_t)gsrc;
    v4u g0;
    g0.x = 1u;                                              // count=1, user desc
    g0.y = lds_byte_addr;                                   // LDS dest (bytes)
    g0.z = (unsigned)(ga & 0xFFFFFFFFull);                  // global_addr[31:0]
    g0.w = (unsigned)((ga >> 32) & 0x01FFFFFFull)           // global_addr[56:32]
         | (2u << 30);                                      // type = 2 ("image")
    v8i g1;
    g1[0] = (int)(2u << 16);         // wg_mask=0, data_size=2 (4 bytes)
    g1[1] = (int)(1024u << 16);      // tensor_dim0[15:0] = 1024 (row length)
    g1[2] = (int)(rows  << 16);      // tensor_dim1[15:0] = rows
    g1[3] = (int)(1024u << 16);      // tile_dim0 = 1024
    g1[4] = (int)rows;               // tile_dim1 = rows, tile_dim2 = 0
    g1[5] = 1024;                    // tensor_dim0_stride = 1024 elements
    g1[6] = 0;
    g1[7] = 0;
    v4i z4 = {};                     // groups 2/3 unused (2-D tensor)
    v8i z8 = {};
    __builtin_amdgcn_tensor_load_to_lds(g0, g1, z4, z4, z8, 0);
}

// ---------------------------------------------------------------------------
// Phase 1: per-token fused kernel.
// One workgroup (256 threads, 8 wave32) per token. The 64x1024 fp32 key tile
// (256 KB) is DMA'd into LDS by the Tensor Data Mover in two 128 KB chunks
// (double-buffered on TENSORcnt) -- no VGPR round-trip, keys read from HBM
// exactly once (1 GiB total). Then: bandwidth, softmax weights, scatter-add
// into vocab probs, weighted key sum, and the X = [query | wsk] GEMM row.
// ---------------------------------------------------------------------------
__global__ __launch_bounds__(256) void ksi_token_kernel(
    const float* __restrict__ query,      // [BT, D]
    const float* __restrict__ keys,       // [BT, K, D]
    const float* __restrict__ distances,  // [BT, K]
    const int*   __restrict__ values,     // [BT, K]
    const float* __restrict__ Wb,         // [2D]
    const float* __restrict__ bb,         // [1]
    float*       __restrict__ out_probs,  // [BT, V] (pre-zeroed)
    float*       __restrict__ X)          // [BT, 2D] workspace
{
    __shared__ float sKeys[KK_ * DD_];    // 256 KB key tile (320 KB LDS/WGP)
    __shared__ float sRed[8];
    __shared__ float sW[KK_];
    __shared__ float sBw;

    const int token = blockIdx.x;
    const int tid   = threadIdx.x;
    const int lane  = tid & 31;
    const int wave  = tid >> 5;

    const float* kbase = keys + (size_t)token * (KK_ * DD_);
    const unsigned ldsKeys = (unsigned)(uintptr_t)&sKeys[0];

    // ---- TDM: issue both 32-row chunks, wait for chunk 0 only -------------
    if (tid == 0) {
        tdm_load_rows_1024f(kbase,               ldsKeys,           32u);
        tdm_load_rows_1024f(kbase + 32 * DD_,    ldsKeys + 131072u, 32u);
        __builtin_amdgcn_s_wait_tensorcnt((short)1);   // chunk 0 landed
    }
    __syncthreads();

    // ---- Pass A: keys . Wb_bot from LDS (chunk 0 now, chunk 1 in flight) --
    float acc = 0.0f;
    const float4* sK4 = (const float4*)sKeys;
    for (int i = tid; i < 8192; i += 256) {            // chunk 0: 32 rows
        float4 v = sK4[i];
        const int d = (i * 4) & (DD_ - 1);
        acc += v.x * Wb[DD_ + d]     + v.y * Wb[DD_ + d + 1]
             + v.z * Wb[DD_ + d + 2] + v.w * Wb[DD_ + d + 3];
    }
    if (tid == 0) __builtin_amdgcn_s_wait_tensorcnt((short)0);  // chunk 1 landed
    __syncthreads();
    for (int i = 8192 + tid; i < 16384; i += 256) {    // chunk 1: rows 32..63
        float4 v = sK4[i];
        const int d = (i * 4) & (DD_ - 1);
        acc += v.x * Wb[DD_ + d]     + v.y * Wb[DD_ + d + 1]
             + v.z * Wb[DD_ + d + 2] + v.w * Wb[DD_ + d + 3];
    }

    // query . Wb_top
    float qacc = 0.0f;
    const float* qrow = query + (size_t)token * DD_;
    for (int d = tid; d < DD_; d += 256) qacc += qrow[d] * Wb[d];

    float part = acc * (1.0f / (float)KK_) + qacc;     // avg = sum/K
    #pragma unroll
    for (int off = 16; off > 0; off >>= 1) part += __shfl_xor(part, off, 32);
    if (lane == 0) sRed[wave] = part;
    __syncthreads();
    if (tid == 0) {
        float s = 0.0f;
        #pragma unroll
        for (int w = 0; w < 8; ++w) s += sRed[w];
        sBw = __expf(s + bb[0]);
    }
    __syncthreads();
    const float bw = sBw;

    // ---- Softmax over K=64 laplacian-kernel scores ------------------------
    if (tid < KK_)
        sW[tid] = -sqrtf(distances[(size_t)token * KK_ + tid]) / bw;
    __syncthreads();
    if (tid == 0) {
        float m = -3.4e38f;
        for (int k = 0; k < KK_; ++k) m = fmaxf(m, sW[k]);
        float s = 0.0f;
        for (int k = 0; k < KK_; ++k) { float e = __expf(sW[k] - m); sW[k] = e; s += e; }
        const float inv = 1.0f / s;
        for (int k = 0; k < KK_; ++k) sW[k] *= inv;
    }
    __syncthreads();

    // ---- Scatter weights into vocab (duplicates possible -> atomic) ------
    if (tid < KK_) {
        const int v = values[(size_t)token * KK_ + tid];
        atomicAdd(&out_probs[(size_t)token * VV_ + v], sW[tid]);
    }

    // ---- Pass B: weighted key sum from LDS; emit X row --------------------
    float* Xrow = X + (size_t)token * D2_;
    for (int d = tid; d < DD_; d += 256) {
        float s = 0.0f;
        #pragma unroll 8
        for (int k = 0; k < KK_; ++k) s += sW[k] * sKeys[k * DD_ + d];
        Xrow[DD_ + d] = s;
        Xrow[d]       = qrow[d];
    }
}

// ---------------------------------------------------------------------------
// Phase 2: H = relu(X @ W1 + b1) via V_WMMA_F32_16X16X4_F32 (fp32 end-to-end).
// X: [4096, 2048], W1: [2048, 1024] row-major, H: [4096, 1024].
// Each wave32 owns a 32x32 C macro-tile (4 WMMA accumulators), k-steps of 4.
//
// f32 WMMA lane layouts (ISA 7.12.2):
//   A 16x4:  lane m = L%16, kpair = (L/16)*2 -> {A[m][kp], A[m][kp+1]}
//   B 4x16:  lane n = L%16, kpair = (L/16)*2 -> {B[kp][n], B[kp+1][n]}
//   C 16x16: vgpr j, lane L -> m = j + 8*(L/16), n = L%16
// ---------------------------------------------------------------------------
__global__ __launch_bounds__(256) void ksi_gemm_kernel(
    const float* __restrict__ X,   // [BT, 2D]
    const float* __restrict__ W1,  // [2D, D]
    const float* __restrict__ b1,  // [D]
    float*       __restrict__ H)   // [BT, D]
{
    const int lane  = threadIdx.x & 31;
    const int wave  = threadIdx.x >> 5;
    const int gwave = blockIdx.x * 8 + wave;      // 4096 waves total
    const int NT    = DD_ / 32;                   // 32 macro-cols
    const int mBase = (gwave / NT) * 32;
    const int nBase = (gwave % NT) * 32;

    const int mA    = lane & 15;
    const int nB    = lane & 15;
    const int kHalf = (lane >> 4) * 2;            // 0 or 2

    v8f c00 = {}, c01 = {}, c10 = {}, c11 = {};

    const float* Arow0 = X + (size_t)(mBase + mA) * D2_;
    const float* Arow1 = X + (size_t)(mBase + 16 + mA) * D2_;

    for (int k = 0; k < D2_; k += 4) {
        const int kk = k + kHalf;
        v2f a0, a1, b0, b1v;
        a0.x = Arow0[kk];     a0.y = Arow0[kk + 1];
        a1.x = Arow1[kk];     a1.y = Arow1[kk + 1];
        const float* Bp = W1 + (size_t)kk * DD_ + nBase + nB;
        b0.x  = Bp[0];        b0.y  = Bp[DD_];
        b1v.x = Bp[16];       b1v.y = Bp[16 + DD_];

        c00 = __builtin_amdgcn_wmma_f32_16x16x4_f32(false, a0, false, b0,  (short)0, c00, false, false);
        c01 = __builtin_amdgcn_wmma_f32_16x16x4_f32(false, a0, false, b1v, (short)0, c01, false, false);
        c10 = __builtin_amdgcn_wmma_f32_16x16x4_f32(false, a1, false, b0,  (short)0, c10, false, false);
        c11 = __builtin_amdgcn_wmma_f32_16x16x4_f32(false, a1, false, b1v, (short)0, c11, false, false);
    }

    // Epilogue: bias + relu, write H
    const int half = lane >> 4;
    const int nn   = lane & 15;
    #pragma unroll
    for (int j = 0; j < 8; ++j) {
        const int m0 = mBase + j + half * 8;
        const int n0 = nBase + nn;
        const float bn0  = b1[n0];
        const float bn16 = b1[n0 + 16];
        H[(size_t)m0        * DD_ + n0     ] = fmaxf(c00[j] + bn0,  0.0f);
        H[(size_t)(m0 + 16) * DD_ + n0     ] = fmaxf(c10[j] + bn0,  0.0f);
        H[(size_t)m0        * DD_ + n0 + 16] = fmaxf(c01[j] + bn16, 0.0f);
        H[(size_t)(m0 + 16) * DD_ + n0 + 16] = fmaxf(c11[j] + bn16, 0.0f);
    }
}

// ---------------------------------------------------------------------------
// Phase 3: lambda = sigmoid(H @ W2 + b2). One wave32 per token, 1024-dot,
// shuffle tree reduction.
// ---------------------------------------------------------------------------
__global__ __launch_bounds__(256) void ksi_lambda_kernel(
    const float* __restrict__ H,   // [BT, D]
    const float* __restrict__ W2,  // [D]
    const float* __restrict__ b2,  // [1]
    float*       __restrict__ outLam)  // [BT]
{
    const int token = blockIdx.x * 8 + (threadIdx.x >> 5);
    const int lane  = threadIdx.x & 31;
    const float* h  = H + (size_t)token * DD_;

    float acc = 0.0f;
    for (int d = lane * 4; d < DD_; d += 128) {
        float4 hv = *(const float4*)(h + d);
        float4 wv = *(const float4*)(W2 + d);
        acc += hv.x * wv.x + hv.y * wv.y + hv.z * wv.z + hv.w * wv.w;
    }
    #pragma unroll
    for (int off = 16; off > 0; off >>= 1) acc += __shfl_xor(acc, off, 32);
    if (lane == 0)
        outLam[token] = 1.0f / (1.0f + __expf(-(acc + b2[0])));
}

// ---------------------------------------------------------------------------
extern "C" void kernel_launch(void* const* d_in, const int* in_sizes, int n_in,
                              void* d_out, int out_size, void* d_ws, size_t ws_size,
                              hipStream_t stream) {
    const float* query     = (const float*)d_in[0];
    const float* keys      = (const float*)d_in[1];
    const float* distances = (const float*)d_in[2];
    const int*   values    = (const int*)  d_in[3];
    const float* Wb        = (const float*)d_in[4];
    const float* bb        = (const float*)d_in[5];
    const float* W1        = (const float*)d_in[6];
    const float* b1        = (const float*)d_in[7];
    const float* W2        = (const float*)d_in[8];
    const float* b2        = (const float*)d_in[9];

    float* out    = (float*)d_out;
    float* X      = (float*)d_ws;                      // [4096 x 2048] = 32 MB
    float* H      = X + (size_t)BT_ * D2_;             // [4096 x 1024] = 16 MB
    float* outLam = out + (size_t)BT_ * VV_;           // lambda tail of d_out

    // Zero the dense [B,T,V] probability output (scatter target).
    (void)hipMemsetAsync(d_out, 0, (size_t)out_size * sizeof(float), stream);

    // Phase 1: one workgroup per token (4096), keys DMA'd to LDS by the TDM.
    ksi_token_kernel<<<BT_, 256, 0, stream>>>(query, keys, distances, values,
                                              Wb, bb, out, X);

    // Phase 2: fp32 WMMA GEMM, 4096 waves = 512 blocks x 8 waves.
    ksi_gemm_kernel<<<512, 256, 0, stream>>>(X, W1, b1, H);

    // Phase 3: lambda GEMV, 8 tokens per block.
    ksi_lambda_kernel<<<BT_ / 8, 256, 0, stream>>>(H, W2, b2, outLam);
}